// CustomCosineEmbeddingLoss_87978110091718
// MI455X (gfx1250) — compile-verified
//
#include <hip/hip_runtime.h>
#include <math.h>

typedef __attribute__((ext_vector_type(2))) float v2f;
typedef __attribute__((ext_vector_type(8))) float v8f;

// Pointer types matching the async-to-LDS builtin's parameters
// (param1 diagnosed as: "int __vector(4) __device__ *"  == v4i addrspace(1)*)
typedef int v4i __attribute__((vector_size(16)));
typedef __attribute__((address_space(1))) v4i* as1_v4i_ptr;
typedef __attribute__((address_space(3))) v4i* as3_v4i_ptr;

#define FEAT_D 512   // feature dim
#define GROWS  16    // rows per wave-group (two 8-row chunks)
#define PANEL  64    // columns staged per LDS panel
#define PSTR   68    // padded row stride in LDS (floats) -> conflict-free b64 reads
#define WPB    8     // waves per block

// ---- CDNA5 async global->LDS path (ASYNCcnt), with safe fallback ----------
#if defined(__gfx1250__) && __has_builtin(__builtin_amdgcn_global_load_async_to_lds_b128)
#define HAVE_ASYNC_LDS 1
#else
#define HAVE_ASYNC_LDS 0
#endif

#if HAVE_ASYNC_LDS
#if __has_builtin(__builtin_amdgcn_s_wait_asynccnt)
#define WAIT_ASYNC(n) __builtin_amdgcn_s_wait_asynccnt(n)
#else
#define WAIT_ASYNC(n) asm volatile("s_wait_asynccnt %0" ::"i"(n) : "memory")
#endif
#else
#define WAIT_ASYNC(n) ((void)0)
#endif

// Stage one 16x64 f32 panel (two rows per pass, b128 per lane, coalesced).
__device__ __forceinline__ void issue_panel(const float* __restrict__ src,
                                            float* __restrict__ buf,
                                            int p, int half, int l16) {
#pragma unroll
  for (int rr = 0; rr < GROWS; rr += 2) {
    const int row = rr + half;
    const float* g = src + (size_t)row * FEAT_D + p + l16 * 4;
    float* l = buf + row * PSTR + l16 * 4;
#if HAVE_ASYNC_LDS
    __builtin_amdgcn_global_load_async_to_lds_b128(
        (as1_v4i_ptr)g, (as3_v4i_ptr)l, 0, 0);
#else
    *reinterpret_cast<float4*>(l) = *reinterpret_cast<const float4*>(g);
#endif
  }
}

// ---------------------------------------------------------------------------
// Kernel 1: per-16-row-group Gram via V_WMMA_F32_16X16X4_F32, cosine loss.
// One wave = one group; double-buffered async LDS panels.
// ---------------------------------------------------------------------------
__global__ void gram_loss_kernel(const float* __restrict__ feat,
                                 float* __restrict__ partial,
                                 int ngroups) {
  extern __shared__ float lds[];            // WPB * 2 * GROWS * PSTR floats
  __shared__ float red[WPB];

  const int lane = threadIdx.x & 31;
  const int wave = threadIdx.x >> 5;
  const int group = blockIdx.x * WPB + wave;

  float waveLoss = 0.0f;
  if (group < ngroups) {                    // wave-uniform; EXEC all-ones inside
    float* buf0 = lds + wave * (2 * GROWS * PSTR);
    float* buf1 = buf0 + GROWS * PSTR;
    const float* src = feat + (size_t)group * (GROWS * FEAT_D);
    const int half = lane >> 4;             // 0: lanes 0-15, 1: lanes 16-31
    const int l16  = lane & 15;

    v8f acc = {};                           // 16x16 f32 accumulator (2 chunk Grams)

    issue_panel(src, buf0, 0, half, l16);   // prologue: panel 0 in flight

#pragma unroll
    for (int p = 0; p < FEAT_D; p += PANEL) {
      float* cur = ((p / PANEL) & 1) ? buf1 : buf0;
      float* nxt = ((p / PANEL) & 1) ? buf0 : buf1;
      if (p + PANEL < FEAT_D) {
        issue_panel(src, nxt, p + PANEL, half, l16);  // prefetch next panel
        WAIT_ASYNC(8);                                // current panel landed
      } else {
        WAIT_ASYNC(0);
      }
      // ---- 16 WMMA K=4 steps over the panel, 4 operand pairs in flight ----
      // A 16x4 layout: lane<16 holds (row=l16, K..K+1); lane>=16 holds K+2..K+3.
      // Symmetric B layout is lane-identical, so 'ab' feeds both operands.
      const float* rowp = cur + l16 * PSTR + half * 2;
#pragma unroll
      for (int k = 0; k < PANEL; k += 16) {
        v2f ab[4];
#pragma unroll
        for (int j = 0; j < 4; ++j) {       // 4 independent ds_load_2addr_b32
          ab[j][0] = rowp[k + j * 4 + 0];
          ab[j][1] = rowp[k + j * 4 + 1];
        }
#pragma unroll
        for (int j = 0; j < 4; ++j) {       // retire with partial dscnt waits
          acc = __builtin_amdgcn_wmma_f32_16x16x4_f32(
              false, ab[j], false, ab[j], (short)0, acc, false, false);
        }
      }
    }

    // ---- epilogue: per-chunk loss = 32 - q^T G q / 2,  q_i = rsqrt(G_ii) ----
    // C/D layout: VGPR j, lanes 0-15 -> G[j][lane]; lanes 16-31 -> G[j+8][lane-16].
    // Chunk0 block at lanes 0-7, chunk1 block at lanes 24-31.
    float r0[8], r1[8];
#pragma unroll
    for (int i = 0; i < 8; ++i) {
      const float t = __shfl(acc[i], i, 32);       // G0[i][i], broadcast
      const float u = __shfl(acc[i], 24 + i, 32);  // G1[i][i], broadcast
      r0[i] = 1.0f / sqrtf(t);
      r1[i] = 1.0f / sqrtf(u);
    }
    float p0 = 0.0f, p1 = 0.0f;                    // row-weighted column sums
#pragma unroll
    for (int i = 0; i < 8; ++i) {
      p0 += r0[i] * acc[i];                        // valid on lanes 0-7
      p1 += r1[i] * acc[i];                        // valid on lanes 24-31
    }
    float tot = 0.0f;
#pragma unroll
    for (int i = 0; i < 8; ++i) {
      tot += r0[i] * __shfl(p0, i, 32);            // q0^T G0 q0
      tot += r1[i] * __shfl(p1, 24 + i, 32);       // q1^T G1 q1
    }
    waveLoss = 64.0f - 0.5f * tot;                 // two chunks: 2*32 - sum/2
  }

  if (lane == 0) red[wave] = waveLoss;
  __syncthreads();
  if (threadIdx.x == 0) {
    float s = 0.0f;
#pragma unroll
    for (int i = 0; i < WPB; ++i) s += red[i];
    partial[blockIdx.x] = s;
  }
}

// ---------------------------------------------------------------------------
// Kernel 2: MSE partial sums, float4 streaming.
// ---------------------------------------------------------------------------
__global__ void mse_kernel(const float* __restrict__ pred,
                           const float* __restrict__ tgt,
                           float* __restrict__ partial, int n4) {
  const int lane = threadIdx.x & 31;
  const int wave = threadIdx.x >> 5;
  const float4* a4 = reinterpret_cast<const float4*>(pred);
  const float4* b4 = reinterpret_cast<const float4*>(tgt);
  float sum = 0.0f;
  for (int i = blockIdx.x * blockDim.x + threadIdx.x; i < n4;
       i += gridDim.x * blockDim.x) {
    const float4 a = a4[i];
    const float4 b = b4[i];
    const float dx = a.x - b.x, dy = a.y - b.y, dz = a.z - b.z, dw = a.w - b.w;
    sum += dx * dx + dy * dy + dz * dz + dw * dw;
  }
#pragma unroll
  for (int off = 16; off > 0; off >>= 1) sum += __shfl_down(sum, off, 32);
  __shared__ float red[8];
  if (lane == 0) red[wave] = sum;
  __syncthreads();
  if (threadIdx.x == 0) {
    float s = 0.0f;
    const int nw = blockDim.x >> 5;
    for (int i = 0; i < nw; ++i) s += red[i];
    partial[blockIdx.x] = s;
  }
}

// ---------------------------------------------------------------------------
// Kernel 3: deterministic final reduction -> scalar output.
// ---------------------------------------------------------------------------
__global__ void final_kernel(const float* __restrict__ gramPart, int ng,
                             const float* __restrict__ msePart, int nm,
                             float* __restrict__ out, float invNC, float alpha) {
  __shared__ float redA[256];
  __shared__ float redB[256];
  float sg = 0.0f, sm = 0.0f;
  for (int i = threadIdx.x; i < ng; i += blockDim.x) sg += gramPart[i];
  for (int i = threadIdx.x; i < nm; i += blockDim.x) sm += msePart[i];
  redA[threadIdx.x] = sg;
  redB[threadIdx.x] = sm;
  __syncthreads();
  for (int s = blockDim.x >> 1; s > 0; s >>= 1) {
    if ((int)threadIdx.x < s) {
      redA[threadIdx.x] += redA[threadIdx.x + s];
      redB[threadIdx.x] += redB[threadIdx.x + s];
    }
    __syncthreads();
  }
  if (threadIdx.x == 0) out[0] = redB[0] * invNC + alpha * redA[0];
}

// ---------------------------------------------------------------------------
extern "C" void kernel_launch(void* const* d_in, const int* in_sizes, int n_in,
                              void* d_out, int out_size, void* d_ws, size_t ws_size,
                              hipStream_t stream) {
  const float* y_pred = (const float*)d_in[0];   // [N, 64]
  const float* y_feat = (const float*)d_in[1];   // [N, 512]
  const float* y_true = (const float*)d_in[2];   // [B,T,64] flat == [N,64]

  const int NC = in_sizes[0];                    // N*C
  const int ND = in_sizes[1];                    // N*D
  const int N  = ND / FEAT_D;
  const int ngroups = N / GROWS;                 // 8192 for the reference shapes
  const int gblocks = (ngroups + WPB - 1) / WPB; // 1024
  const int mblocks = 1024;

  float* ws = (float*)d_ws;
  float* gramPart = ws;
  float* msePart  = ws + gblocks;

  const size_t shmem = (size_t)WPB * 2 * GROWS * PSTR * sizeof(float); // 69632 B

  gram_loss_kernel<<<gblocks, WPB * 32, shmem, stream>>>(y_feat, gramPart, ngroups);
  mse_kernel<<<mblocks, 256, 0, stream>>>(y_pred, y_true, msePart, NC / 4);
  final_kernel<<<1, 256, 0, stream>>>(gramPart, gblocks, msePart, mblocks,
                                      (float*)d_out, 1.0f / (float)NC, 0.1f);
}